// LegacyMHA_79740362817871
// MI455X (gfx1250) — compile-verified
//
#include <hip/hip_runtime.h>

#define B_  8
#define Q_  16
#define L_  4096
#define E_  1024
#define LK  (L_ + Q_)   // 4112

typedef __attribute__((ext_vector_type(2))) float v2f;
typedef __attribute__((ext_vector_type(8))) float v8f;

__device__ __forceinline__ v8f wmma_f32(v2f a, v2f b, v8f c) {
  // D = A(16x4,f32) x B(4x16,f32) + C(16x16,f32)
  return __builtin_amdgcn_wmma_f32_16x16x4_f32(
      /*neg_a=*/false, a, /*neg_b=*/false, b,
      /*c_mod=*/(short)0, c, /*reuse_a=*/false, /*reuse_b=*/false);
}

// ---------------------------------------------------------------------------
// 1) KV-cache concat copy: k_cache/v_cache -> K[:, :L, :], V[:, :L, :]
//    Pure bandwidth: float4 streaming copy, per-batch contiguous chunks.
// ---------------------------------------------------------------------------
__global__ __launch_bounds__(256)
void copy_cache_kernel(const float4* __restrict__ kc,
                       const float4* __restrict__ vc,
                       float4* __restrict__ Kout,
                       float4* __restrict__ Vout) {
  const size_t per_batch = (size_t)L_ * E_ / 4;   // 1,048,576 float4
  size_t i = (size_t)blockIdx.x * blockDim.x + threadIdx.x;
  if (i >= per_batch) return;
  const int b = blockIdx.y;
  const size_t dst_stride = (size_t)LK * E_ / 4;
  if (blockIdx.z == 0)
    Kout[(size_t)b * dst_stride + i] = kc[(size_t)b * per_batch + i];
  else
    Vout[(size_t)b * dst_stride + i] = vc[(size_t)b * per_batch + i];
}

// ---------------------------------------------------------------------------
// 2) Projections: X[128,1024] @ W^T + bias for W in {Wq,Wk,Wv}
//    One wave per 16x16 output tile. q -> workspace; k,v -> concat tail of K,V.
// ---------------------------------------------------------------------------
__global__ __launch_bounds__(32)
void proj_kernel(const float* __restrict__ query,
                 const float* __restrict__ Wq, const float* __restrict__ bq,
                 const float* __restrict__ Wk, const float* __restrict__ bk,
                 const float* __restrict__ Wv, const float* __restrict__ bv,
                 float* __restrict__ q_buf,
                 float* __restrict__ Kout,
                 float* __restrict__ Vout) {
  const int lane = threadIdx.x;        // 0..31, full wave32
  const int n0   = blockIdx.x * 16;    // output feature tile
  const int m0   = blockIdx.y * 16;    // row tile over B*Q
  const int proj = blockIdx.z;         // 0=q 1=k 2=v

  const float* W    = (proj == 0) ? Wq : (proj == 1) ? Wk : Wv;
  const float* bias = (proj == 0) ? bq : (proj == 1) ? bk : bv;

  const int lm = lane & 15;
  const int lh = lane >> 4;            // 0|1 selects K pair {0,1} vs {2,3}
  const float* arow = query + (size_t)(m0 + lm) * E_;
  const float* brow = W     + (size_t)(n0 + lm) * E_;   // B[k,n] = W[n,k]

  v8f acc = {};
#pragma unroll 4
  for (int k0 = 0; k0 < E_; k0 += 4) {
    const int ka = k0 + 2 * lh;
    v2f a = *(const v2f*)(arow + ka);
    v2f b = *(const v2f*)(brow + ka);
    acc = wmma_f32(a, b, acc);
  }

  const float bn = bias[n0 + lm];
#pragma unroll
  for (int v = 0; v < 8; ++v) {
    const int m = m0 + v + 8 * lh;     // global row in [0,128)
    const int n = n0 + lm;
    const float val = acc[v] + bn;
    if (proj == 0) {
      q_buf[(size_t)m * E_ + n] = val;
    } else {
      const int b  = m >> 4;           // batch
      const int qi = m & 15;           // query index
      float* dst = (proj == 1) ? Kout : Vout;
      dst[((size_t)b * LK + L_ + qi) * E_ + n] = val;
    }
  }
}

// ---------------------------------------------------------------------------
// 3) scores[b, 0:16, n0:n0+16] = (q_b @ K_b^T) * 1/sqrt(E)
// ---------------------------------------------------------------------------
__global__ __launch_bounds__(32)
void scores_kernel(const float* __restrict__ q_buf,
                   const float* __restrict__ Kmat,
                   float* __restrict__ scores) {
  const int lane = threadIdx.x;
  const int n0 = blockIdx.x * 16;      // key tile (LK/16 = 257 exact)
  const int b  = blockIdx.y;
  const int lm = lane & 15;
  const int lh = lane >> 4;

  const float* arow = q_buf + ((size_t)b * Q_ + lm) * E_;
  const float* brow = Kmat  + ((size_t)b * LK + n0 + lm) * E_;  // B[k,n]=K[n,k]

  v8f acc = {};
#pragma unroll 4
  for (int k0 = 0; k0 < E_; k0 += 4) {
    const int ka = k0 + 2 * lh;
    v2f a  = *(const v2f*)(arow + ka);
    v2f bb = *(const v2f*)(brow + ka);
    acc = wmma_f32(a, bb, acc);
  }

  const float scale = 0.03125f;        // 1/sqrt(1024)
#pragma unroll
  for (int v = 0; v < 8; ++v) {
    const int m = v + 8 * lh;
    scores[((size_t)b * Q_ + m) * LK + n0 + lm] = acc[v] * scale;
  }
}

// ---------------------------------------------------------------------------
// 4) softmax over each of the 128 rows of length LK (in place)
// ---------------------------------------------------------------------------
__global__ __launch_bounds__(256)
void softmax_kernel(float* __restrict__ scores) {
  const int row = blockIdx.x;          // 0..B*Q-1
  float* p = scores + (size_t)row * LK;
  __shared__ float red[256];
  const int t = threadIdx.x;

  float m = -3.402823466e38f;
  for (int i = t; i < LK; i += 256) m = fmaxf(m, p[i]);
  red[t] = m; __syncthreads();
  for (int s = 128; s > 0; s >>= 1) {
    if (t < s) red[t] = fmaxf(red[t], red[t + s]);
    __syncthreads();
  }
  m = red[0]; __syncthreads();

  float sum = 0.f;
  for (int i = t; i < LK; i += 256) {
    const float e = __expf(p[i] - m);
    p[i] = e;
    sum += e;
  }
  red[t] = sum; __syncthreads();
  for (int s = 128; s > 0; s >>= 1) {
    if (t < s) red[t] += red[t + s];
    __syncthreads();
  }
  const float inv = 1.0f / red[0];
  for (int i = t; i < LK; i += 256) p[i] *= inv;
}

// ---------------------------------------------------------------------------
// 5) out[b, 0:16, n0:n0+16] = attn_b @ V_b   (K-dim = LK = 4112 = 4*1028)
//    V reads are naturally row-major -> fully coalesced B fragments.
// ---------------------------------------------------------------------------
__global__ __launch_bounds__(32)
void av_kernel(const float* __restrict__ attn,
               const float* __restrict__ Vmat,
               float* __restrict__ out) {
  const int lane = threadIdx.x;
  const int n0 = blockIdx.x * 16;      // embedding tile
  const int b  = blockIdx.y;
  const int lm = lane & 15;
  const int lh = lane >> 4;

  const float* arow = attn + ((size_t)b * Q_ + lm) * LK;
  const float* Vb   = Vmat + (size_t)b * LK * E_ + n0 + lm;

  v8f acc = {};
#pragma unroll 4
  for (int k0 = 0; k0 < LK; k0 += 4) {
    const int ka = k0 + 2 * lh;
    v2f a = *(const v2f*)(arow + ka);
    v2f bb;
    bb.x = Vb[(size_t)ka * E_];
    bb.y = Vb[(size_t)(ka + 1) * E_];
    acc = wmma_f32(a, bb, acc);
  }

#pragma unroll
  for (int v = 0; v < 8; ++v) {
    const int m = v + 8 * lh;
    out[((size_t)b * Q_ + m) * E_ + n0 + lm] = acc[v];
  }
}

// ---------------------------------------------------------------------------
extern "C" void kernel_launch(void* const* d_in, const int* in_sizes, int n_in,
                              void* d_out, int out_size, void* d_ws, size_t ws_size,
                              hipStream_t stream) {
  const float* query   = (const float*)d_in[0];
  const float* k_cache = (const float*)d_in[1];
  const float* v_cache = (const float*)d_in[2];
  const float* Wq = (const float*)d_in[3];
  const float* bq = (const float*)d_in[4];
  const float* Wk = (const float*)d_in[5];
  const float* bk = (const float*)d_in[6];
  const float* Wv = (const float*)d_in[7];
  const float* bv = (const float*)d_in[8];

  // d_out = [out (B*Q*E) | K (B*LK*E) | V (B*LK*E)]
  float* out  = (float*)d_out;
  float* Kout = out  + (size_t)B_ * Q_ * E_;
  float* Vout = Kout + (size_t)B_ * LK * E_;

  // workspace: q projections + score matrix (~2.6 MB)
  float* q_buf  = (float*)d_ws;                       // B*Q*E floats
  float* scores = q_buf + (size_t)B_ * Q_ * E_;       // B*Q*LK floats

  // 1) stream the KV cache into the concat outputs
  copy_cache_kernel<<<dim3((L_ * E_ / 4) / 256, B_, 2), 256, 0, stream>>>(
      (const float4*)k_cache, (const float4*)v_cache,
      (float4*)Kout, (float4*)Vout);

  // 2) q/k/v projections (k,v land in the concat tails of K,V)
  proj_kernel<<<dim3(E_ / 16, (B_ * Q_) / 16, 3), 32, 0, stream>>>(
      query, Wq, bq, Wk, bk, Wv, bv, q_buf, Kout, Vout);

  // 3) attention scores (needs full K incl. projected tail)
  scores_kernel<<<dim3(LK / 16, B_), 32, 0, stream>>>(q_buf, Kout, scores);

  // 4) softmax rows
  softmax_kernel<<<B_ * Q_, 256, 0, stream>>>(scores);

  // 5) attn @ V
  av_kernel<<<dim3(E_ / 16, B_), 32, 0, stream>>>(scores, Vout, out);
}